// Fixed_pool_81277961109623
// MI455X (gfx1250) — compile-verified
//
#include <hip/hip_runtime.h>

// 2x2 pixel-unshuffle (lazy wavelet split) for x:(8,256,256,256) f32.
// Outputs ll,lh,hl,hh each (8,256,128,128), concatenated flat in d_out.
//
// Pure streaming kernel: 1 GiB total traffic, ~46us floor at 23.3 TB/s.
// Strategy: b128 loads/stores only, lane-local even/odd deinterleave,
// non-temporal cache hints since both streams exceed the 192MB L2 and
// are touched exactly once.

typedef float float4v __attribute__((ext_vector_type(4)));

__global__ __launch_bounds__(256) void lazy_wavelet_split(
    const float* __restrict__ x,
    float* __restrict__ out,
    unsigned int n_groups,   // total 8-float groups = N/8
    unsigned int quarter)    // elements per output quarter = N/4
{
    unsigned int idx = blockIdx.x * blockDim.x + threadIdx.x;
    if (idx >= n_groups) return;

    // Each thread owns 8 consecutive floats of one input row (W = 256).
    // 256/8 = 32 groups per row.
    unsigned int t      = idx & 31u;   // group within row -> out col base = 4t
    unsigned int row    = idx >> 5;    // row index over (B*C*H)
    unsigned int h      = row & 255u;  // H = 256
    unsigned int plane  = row >> 8;    // (B*C) plane index
    unsigned int h_out  = h >> 1;
    unsigned int parity = h & 1u;      // 0 -> ll/lh, 1 -> hl/hh

    const float4v* in = (const float4v*)(x + (size_t)idx * 8u);
    float4v a = __builtin_nontemporal_load(in);      // cols 8t .. 8t+3
    float4v b = __builtin_nontemporal_load(in + 1);  // cols 8t+4 .. 8t+7

    float4v even = { a.x, a.z, b.x, b.z };  // cols 8t,8t+2,8t+4,8t+6 -> out 4t..4t+3
    float4v odd  = { a.y, a.w, b.y, b.w };  // cols 8t+1,...        -> out 4t..4t+3

    // Output quarter layout: plane * (128*128) + h_out * 128 + col
    size_t out_off = (size_t)plane * 16384u + (size_t)h_out * 128u + 4u * t;
    float* base_even = out + (size_t)(parity ? 2u : 0u) * quarter;  // ll or hl
    float* base_odd  = out + (size_t)(parity ? 3u : 1u) * quarter;  // lh or hh

    __builtin_nontemporal_store(even, (float4v*)(base_even + out_off));
    __builtin_nontemporal_store(odd,  (float4v*)(base_odd  + out_off));
}

extern "C" void kernel_launch(void* const* d_in, const int* in_sizes, int n_in,
                              void* d_out, int out_size, void* d_ws, size_t ws_size,
                              hipStream_t stream) {
    const float* x = (const float*)d_in[0];
    float* out = (float*)d_out;

    unsigned int n        = (unsigned int)in_sizes[0];  // 8*256*256*256 = 134217728
    unsigned int n_groups = n / 8u;                     // 16777216 threads
    unsigned int quarter  = n / 4u;                     // 33554432 elems per output

    const unsigned int threads = 256;                   // 8 waves (wave32) per block
    unsigned int blocks = (n_groups + threads - 1u) / threads;

    lazy_wavelet_split<<<blocks, threads, 0, stream>>>(x, out, n_groups, quarter);
}